// GAT_13400297963982
// MI455X (gfx1250) — compile-verified
//
#include <hip/hip_runtime.h>
#include <hip/hip_bf16.h>
#include <math.h>

#define NITEM 80000
#define NUSER 20000
#define NN    100000
#define DF    64
#define UD    32
#define NE    2000000

typedef float v2f __attribute__((ext_vector_type(2)));
typedef float v8f __attribute__((ext_vector_type(8)));

// fp32 WMMA: D(16x16) = A(16x4) * B(4x16) + C.  8-arg form:
// (neg_a, A, neg_b, B, c_mod, C, reuse_a, reuse_b)
__device__ __forceinline__ v8f wmma4(v2f a, v2f b, v8f c) {
    return __builtin_amdgcn_wmma_f32_16x16x4_f32(false, a, false, b, (short)0, c,
                                                 false, false);
}

__device__ __forceinline__ float lrelu(float v) { return v > 0.0f ? v : 0.01f * v; }

// ---------------- init scratch (deterministic every launch) ----------------
__global__ void k_init(float* __restrict__ agg, float* __restrict__ denom,
                       float* __restrict__ segmax) {
    size_t i = (size_t)blockIdx.x * blockDim.x + threadIdx.x;
    if (i < (size_t)NN * DF) agg[i] = 0.0f;
    if (i < (size_t)NN) { denom[i] = 0.0f; segmax[i] = -1e30f; }
}

// ---------------- user MLP: tanh(uf @ W_user^T + b) -> x[NITEM..] ----------
// one wave computes 16 rows x 64 cols; K=32 in 8 WMMA k-steps
__global__ void __launch_bounds__(256) k_user_mlp(const float* __restrict__ uf,
                                                  const float* __restrict__ Wu,
                                                  const float* __restrict__ bu,
                                                  float* __restrict__ x) {
    int wave = threadIdx.x >> 5, lane = threadIdx.x & 31;
    int tile = blockIdx.x * (blockDim.x >> 5) + wave;
    if (tile >= NUSER / 16) return;        // wave-uniform: EXEC stays all-ones
    int sub = lane & 15, half = lane >> 4;
    int row0 = tile * 16;

    v8f acc[4] = {};
    for (int kk = 0; kk < UD / 4; ++kk) {
        int k0 = kk * 4 + half * 2;
        const float* ap = uf + (size_t)(row0 + sub) * UD + k0;
        v2f a; a[0] = ap[0]; a[1] = ap[1];
        for (int nt = 0; nt < 4; ++nt) {
            int n = nt * 16 + sub;
            const float* bp = Wu + (size_t)n * UD + k0;  // B[k][n] = W_user[n][k]
            v2f b; b[0] = bp[0]; b[1] = bp[1];
            acc[nt] = wmma4(a, b, acc[nt]);
        }
    }
    for (int nt = 0; nt < 4; ++nt) {
        int col = nt * 16 + sub;
        float bb = bu[col];
        for (int v = 0; v < 8; ++v) {
            int row = row0 + v + half * 8;
            x[(size_t)(NITEM + row) * DF + col] = tanhf(acc[nt][v] + bb);
        }
    }
}

// ---------------- row-wise L2 normalize of x (items from features) ---------
__global__ void __launch_bounds__(256) k_norm(const float* __restrict__ feat,
                                              float* __restrict__ x) {
    int wave = threadIdx.x >> 5, lane = threadIdx.x & 31;
    int row = blockIdx.x * (blockDim.x >> 5) + wave;
    if (row >= NN) return;
    const float* src = (row < NITEM) ? (feat + (size_t)row * DF)
                                     : (x + (size_t)row * DF);
    float a = src[2 * lane], b = src[2 * lane + 1];
    float ss = a * a + b * b;
    for (int off = 1; off < 32; off <<= 1) ss += __shfl_xor(ss, off, 32);
    float inv = 1.0f / fmaxf(sqrtf(ss), 1e-12f);
    x[(size_t)row * DF + 2 * lane]     = a * inv;
    x[(size_t)row * DF + 2 * lane + 1] = b * inv;
}

// ---------------- xw = x @ W_gat  (WMMA, K=64) -----------------------------
__global__ void __launch_bounds__(256) k_xw(const float* __restrict__ x,
                                            const float* __restrict__ Wg,
                                            float* __restrict__ xw) {
    int wave = threadIdx.x >> 5, lane = threadIdx.x & 31;
    int tile = blockIdx.x * (blockDim.x >> 5) + wave;
    if (tile >= NN / 16) return;
    int sub = lane & 15, half = lane >> 4;
    int row0 = tile * 16;

    v8f acc[4] = {};
    for (int kk = 0; kk < DF / 4; ++kk) {
        int k0 = kk * 4 + half * 2;
        const float* ap = x + (size_t)(row0 + sub) * DF + k0;
        v2f a; a[0] = ap[0]; a[1] = ap[1];
        for (int nt = 0; nt < 4; ++nt) {
            int n = nt * 16 + sub;
            v2f b;                                 // B[k][n] = W_gat[k][n]
            b[0] = Wg[(size_t)k0 * DF + n];
            b[1] = Wg[(size_t)(k0 + 1) * DF + n];
            acc[nt] = wmma4(a, b, acc[nt]);
        }
    }
    for (int nt = 0; nt < 4; ++nt) {
        int col = nt * 16 + sub;
        for (int v = 0; v < 8; ++v) {
            int row = row0 + v + half * 8;
            xw[(size_t)row * DF + col] = acc[nt][v];
        }
    }
}

// ---------------- edge pass 1: per-dst max of scores (L2-resident gathers) -
__global__ void __launch_bounds__(256) k_edge_max(const int* __restrict__ srcI,
                                                  const int* __restrict__ dstI,
                                                  const float* __restrict__ xw,
                                                  float* __restrict__ segmax) {
    int wid  = blockIdx.x * (blockDim.x >> 5) + (threadIdx.x >> 5);
    int lane = threadIdx.x & 31, sub = lane & 15, half = lane >> 4;
    int e = wid * 2 + half;                 // one edge per half-wave
    float p = 0.0f; int d = 0; bool ok = false;
    if (e < NE) {
        int s = srcI[e]; d = dstI[e];
        if (s != d) {                        // remove_self_loops
            ok = true;
            const float4* ps = (const float4*)(xw + (size_t)s * DF);
            const float4* pd = (const float4*)(xw + (size_t)d * DF);
            float4 a = ps[sub], b = pd[sub]; // 256B coalesced row reads
            p = a.x * b.x + a.y * b.y + a.z * b.z + a.w * b.w;
        }
    }
    for (int off = 1; off < 16; off <<= 1) p += __shfl_xor(p, off, 32);
    if (ok && sub == 0) atomicMax(segmax + d, p);  // global_atomic_max_num_f32
}

// ---------------- edge pass 2: e=exp(score-max); scatter e and e*xj --------
__global__ void __launch_bounds__(256) k_edge_scatter(const int* __restrict__ srcI,
                                                      const int* __restrict__ dstI,
                                                      const float* __restrict__ xw,
                                                      const float* __restrict__ segmax,
                                                      float* __restrict__ denom,
                                                      float* __restrict__ agg) {
    int wid  = blockIdx.x * (blockDim.x >> 5) + (threadIdx.x >> 5);
    int lane = threadIdx.x & 31, sub = lane & 15, half = lane >> 4;
    int e = wid * 2 + half;
    float p = 0.0f; int d = 0; bool ok = false;
    float4 a = make_float4(0.f, 0.f, 0.f, 0.f);
    if (e < NE) {
        int s = srcI[e]; d = dstI[e];
        if (s != d) {
            ok = true;
            const float4* ps = (const float4*)(xw + (size_t)s * DF);
            const float4* pd = (const float4*)(xw + (size_t)d * DF);
            a = ps[sub];
            float4 b = pd[sub];
            p = a.x * b.x + a.y * b.y + a.z * b.z + a.w * b.w;
        }
    }
    for (int off = 1; off < 16; off <<= 1) p += __shfl_xor(p, off, 32);
    if (ok) {
        float m = segmax[d];
        m = (m > -5e29f) ? m : 0.0f;         // empty-segment convention
        float ex = expf(p - m);
        if (sub == 0) atomicAdd(denom + d, ex);
        float* ap = agg + (size_t)d * DF + sub * 4;
        atomicAdd(ap + 0, ex * a.x);         // global_atomic_add_f32 (no return)
        atomicAdd(ap + 1, ex * a.y);
        atomicAdd(ap + 2, ex * a.z);
        atomicAdd(ap + 3, ex * a.w);
    }
}

// ---- finalize: h = lrelu(l2norm(agg/denom + b_gat)); two fused WMMA GEMMs -
__global__ void __launch_bounds__(256) k_final(const float* __restrict__ x,
                                               const float* __restrict__ agg,
                                               const float* __restrict__ denom,
                                               const float* __restrict__ bgat,
                                               const float* __restrict__ Wlin1,
                                               const float* __restrict__ blin1,
                                               const float* __restrict__ Wg1,
                                               const float* __restrict__ bg1,
                                               const float* __restrict__ idw,
                                               float* __restrict__ out) {
    __shared__ float hbuf[8 * 16 * DF];      // 32 KB: per-wave 16x64 h tile
    int wave = threadIdx.x >> 5, lane = threadIdx.x & 31;
    int tileId = blockIdx.x * 8 + wave;
    const int NT = NN / 16;
    int tile = tileId < NT ? tileId : NT - 1;   // clamp (no early-exit: barrier)
    bool valid = tileId < NT;
    int sub = lane & 15, half = lane >> 4;
    int row0 = tile * 16;
    float* hl = hbuf + wave * 16 * DF;

    for (int r = 0; r < 16; ++r) {
        int row = row0 + r;
        float dn = fmaxf(denom[row], 1e-16f);
        float va = agg[(size_t)row * DF + 2 * lane]     / dn + bgat[2 * lane];
        float vb = agg[(size_t)row * DF + 2 * lane + 1] / dn + bgat[2 * lane + 1];
        float ss = va * va + vb * vb;
        for (int off = 1; off < 32; off <<= 1) ss += __shfl_xor(ss, off, 32);
        float inv = 1.0f / fmaxf(sqrtf(ss), 1e-12f);
        hl[r * DF + 2 * lane]     = lrelu(va * inv);
        hl[r * DF + 2 * lane + 1] = lrelu(vb * inv);
    }
    __syncthreads();

    v8f acc1[4] = {};   // x @ W_lin1^T
    v8f acc2[4] = {};   // h @ W_g1^T
    for (int kk = 0; kk < DF / 4; ++kk) {
        int k0 = kk * 4 + half * 2;
        const float* xp = x + (size_t)(row0 + sub) * DF + k0;
        v2f ax; ax[0] = xp[0]; ax[1] = xp[1];
        v2f ah; ah[0] = hl[sub * DF + k0]; ah[1] = hl[sub * DF + k0 + 1];
        for (int nt = 0; nt < 4; ++nt) {
            int n = nt * 16 + sub;
            const float* w1 = Wlin1 + (size_t)n * DF + k0;  // B[k][n]=W[n][k]
            const float* w2 = Wg1   + (size_t)n * DF + k0;
            v2f b1; b1[0] = w1[0]; b1[1] = w1[1];
            v2f b2; b2[0] = w2[0]; b2[1] = w2[1];
            acc1[nt] = wmma4(ax, b1, acc1[nt]);
            acc2[nt] = wmma4(ah, b2, acc2[nt]);
        }
    }
    if (valid) {
        for (int nt = 0; nt < 4; ++nt) {
            int col = nt * 16 + sub;
            float bl = blin1[col], bg = bg1[col];
            for (int v = 0; v < 8; ++v) {
                int row = row0 + v + half * 8;
                float xhat = lrelu(acc1[nt][v] + bl) + idw[(size_t)row * DF + col];
                out[(size_t)row * DF + col] = lrelu(acc2[nt][v] + bg + xhat);
            }
        }
    }
}

extern "C" void kernel_launch(void* const* d_in, const int* in_sizes, int n_in,
                              void* d_out, int out_size, void* d_ws, size_t ws_size,
                              hipStream_t stream) {
    const float* features      = (const float*)d_in[0];
    const float* user_features = (const float*)d_in[1];
    const float* id_weight     = (const float*)d_in[2];
    const float* W_user        = (const float*)d_in[3];
    const float* b_user        = (const float*)d_in[4];
    const float* W_gat         = (const float*)d_in[5];
    const float* b_gat         = (const float*)d_in[6];
    const float* W_lin1        = (const float*)d_in[7];
    const float* b_lin1        = (const float*)d_in[8];
    const float* W_g1          = (const float*)d_in[9];
    const float* b_g1          = (const float*)d_in[10];
    const int*   ei            = (const int*)d_in[11];
    const int*   srcI = ei;
    const int*   dstI = ei + NE;
    float* out = (float*)d_out;

    float* x      = (float*)d_ws;                   // NN*DF
    float* xw     = x   + (size_t)NN * DF;          // NN*DF
    float* agg    = xw  + (size_t)NN * DF;          // NN*DF
    float* denom  = agg + (size_t)NN * DF;          // NN
    float* segmax = denom + NN;                     // NN

    dim3 blk(256);
    k_init<<<((size_t)NN * DF + 255) / 256, blk, 0, stream>>>(agg, denom, segmax);
    k_user_mlp<<<(NUSER / 16 + 7) / 8, blk, 0, stream>>>(user_features, W_user, b_user, x);
    k_norm<<<(NN + 7) / 8, blk, 0, stream>>>(features, x);
    k_xw<<<(NN / 16 + 7) / 8, blk, 0, stream>>>(x, W_gat, xw);
    int nwaves = NE / 2;   // one edge per half-wave
    k_edge_max<<<(nwaves + 7) / 8, blk, 0, stream>>>(srcI, dstI, xw, segmax);
    k_edge_scatter<<<(nwaves + 7) / 8, blk, 0, stream>>>(srcI, dstI, xw, segmax, denom, agg);
    k_final<<<(NN / 16 + 7) / 8, blk, 0, stream>>>(x, agg, denom, b_gat, W_lin1, b_lin1,
                                                   W_g1, b_g1, id_weight, out);
}